// OFTLayer_20461224198407
// MI455X (gfx1250) — compile-verified
//
#include <hip/hip_runtime.h>
#include <hip/hip_bf16.h>

typedef float v2f __attribute__((ext_vector_type(2)));
typedef float v8f __attribute__((ext_vector_type(8)));

#define OFT_EPS     1e-5f
#define NUM_BLOCKS  1024
#define RANK        4
#define D_FEAT      4096      // 1024 blocks * rank 4
#define N_ROWS      16384     // 4 * 4096

// ---------------------------------------------------------------------------
// Kernel 1: Cayley transform + pack B matrices in WMMA B-operand lane layout.
//
// For block b (g = b & 3 = position inside its 16-feature column group):
//   S  = R - R^T
//   A  = I + S + eps*I
//   Q  = (I - S) * A^{-1}
// WMMA computes D[m][n] = sum_k A[m][k] * B[k][n]; we need
//   out[n, 4b+j] = sum_i x[n,4b+i] * Q[j][i]  =>  B[i][4g + j] = Q[j][i],
// with zeros in the other 12 columns (so 4 chained WMMAs build the 16x16
// block-diagonal product).
//
// B lane layout for V_WMMA_F32_16X16X4_F32 (4x16, 2 VGPRs, wave32):
//   VGPR0: lanes 0-15 -> (K=0, N=lane), lanes 16-31 -> (K=2, N=lane-16)
//   VGPR1: lanes 0-15 -> (K=1, N=lane), lanes 16-31 -> (K=3, N=lane-16)
// Stored interleaved: ws[b*64 + 2*lane + {0,1}] so each lane does one b64 load.
// ---------------------------------------------------------------------------
__global__ void oft_cayley_pack(const float* __restrict__ R,
                                float* __restrict__ Bws) {
    int b = blockIdx.x * blockDim.x + threadIdx.x;
    if (b >= NUM_BLOCKS) return;

    float r[4][4];
#pragma unroll
    for (int i = 0; i < 4; ++i)
#pragma unroll
        for (int j = 0; j < 4; ++j)
            r[i][j] = R[b * 16 + i * 4 + j];

    float S[4][4], M[4][8], Nm[4][4];
#pragma unroll
    for (int i = 0; i < 4; ++i) {
#pragma unroll
        for (int j = 0; j < 4; ++j) {
            float s = r[i][j] - r[j][i];
            S[i][j]  = s;
            M[i][j]  = s + ((i == j) ? (1.0f + OFT_EPS) : 0.0f); // I + S + eps I
            Nm[i][j] = ((i == j) ? 1.0f : 0.0f) - s;             // I - S
            M[i][j + 4] = (i == j) ? 1.0f : 0.0f;                // augmented I
        }
    }

    // Gauss-Jordan inverse of M (diag = 1+eps, skew off-diag ~0.04: no pivot needed)
#pragma unroll
    for (int k = 0; k < 4; ++k) {
        float p = 1.0f / M[k][k];
#pragma unroll
        for (int c = 0; c < 8; ++c) M[k][c] *= p;
#pragma unroll
        for (int i = 0; i < 4; ++i) {
            if (i == k) continue;
            float f = M[i][k];
#pragma unroll
            for (int c = 0; c < 8; ++c) M[i][c] -= f * M[k][c];
        }
    }

    float Q[4][4];
#pragma unroll
    for (int i = 0; i < 4; ++i)
#pragma unroll
        for (int j = 0; j < 4; ++j) {
            float acc = 0.0f;
#pragma unroll
            for (int k = 0; k < 4; ++k) acc += Nm[i][k] * M[k][j + 4];
            Q[i][j] = acc;
        }

    const int g = b & 3;
    float* dst = Bws + (size_t)b * 64;
#pragma unroll
    for (int L = 0; L < 32; ++L) {
        int n  = L & 15;            // column N of the 16-wide group
        int k0 = (L >> 4) * 2;      // K covered by this half-wave: {0,1} or {2,3}
        int j  = n - 4 * g;         // local output index within this block
        float v0 = 0.0f, v1 = 0.0f;
        if (j >= 0 && j < 4) { v0 = Q[j][k0]; v1 = Q[j][k0 + 1]; }
        dst[2 * L + 0] = v0;
        dst[2 * L + 1] = v1;
    }
}

// ---------------------------------------------------------------------------
// Kernel 2: streaming block-diagonal transform via chained WMMA f32 16x16x4.
//
// Wave (32 lanes) owns a 16-feature column group (4 blocks), loads its 4
// packed B operands once (regular temporal policy -> stays in L2, 256 KB
// total vs 192 MB L2), then streams 16 row-tiles of 16 rows:
//   per tile: 4x global_load_b64 NT (A operands, exact WMMA A layout; x is
//             touched exactly once -> non-temporal so the 256 MB stream
//             doesn't evict the reused B operands),
//             4x v_wmma_f32_16x16x4_f32 chained on C (block-diag accumulate),
//             8x b32 NT stores (out is written once, never re-read).
// Traffic = exactly one read + one write of x: the HBM roofline (~23 us at
// 23.3 TB/s for 537 MB).
// ---------------------------------------------------------------------------
__global__ void __launch_bounds__(256)
oft_apply_wmma(const float* __restrict__ x,
               const float* __restrict__ Bws,
               float* __restrict__ out) {
    const int lane = threadIdx.x & 31;
    const int wave = threadIdx.x >> 5;

    const int d0 = blockIdx.x * 128 + wave * 16;   // feature base of 16-wide group
    const int b0 = d0 >> 2;                        // first of this group's 4 blocks

    // Load the 4 pre-packed B operands (one b64 per lane each), cacheable.
    v2f bm[4];
#pragma unroll
    for (int g = 0; g < 4; ++g) {
        const float* p = Bws + (size_t)(b0 + g) * 64 + 2 * lane;
        bm[g] = *(const v2f*)p;
    }

    const int m     = lane & 15;      // A-matrix row M held by this lane
    const int half  = lane >> 4;      // 0: K={0,1}, 1: K={2,3}
    const float* xbase = x   + (size_t)m * D_FEAT + d0 + (half << 1);
    const int row_base = blockIdx.y * 256;

    for (int t = 0; t < 16; ++t) {
        const int n0 = row_base + t * 16;
        const float* xp = xbase + (size_t)n0 * D_FEAT;

        v8f c = {};
#pragma unroll
        for (int g = 0; g < 4; ++g) {
            v2f a = __builtin_nontemporal_load((const v2f*)(xp + 4 * g));
            // (neg_a, A, neg_b, B, c_mod, C, reuse_a, reuse_b)
            c = __builtin_amdgcn_wmma_f32_16x16x4_f32(
                    false, a, false, bm[g], (short)0, c, false, false);
        }

        // D layout: c[rr] = D[M = rr + 8*half][N = m] -> out[n0 + rr + 8*half, d0 + m]
        float* op = out + (size_t)(n0 + (half << 3)) * D_FEAT + d0 + m;
#pragma unroll
        for (int rr = 0; rr < 8; ++rr) {
            __builtin_nontemporal_store(c[rr], op + (size_t)rr * D_FEAT);
        }
    }
}

extern "C" void kernel_launch(void* const* d_in, const int* in_sizes, int n_in,
                              void* d_out, int out_size, void* d_ws, size_t ws_size,
                              hipStream_t stream) {
    const float* x     = (const float*)d_in[0];   // [4, 4096, 4096] fp32
    const float* oft_r = (const float*)d_in[1];   // [1024, 4, 4]   fp32
    float*       out   = (float*)d_out;           // [4, 4096, 4096] fp32
    float*       Bws   = (float*)d_ws;            // 1024 * 64 floats = 256 KB

    // 1) Cayley transform + WMMA B-operand packing (1024 threads, negligible).
    oft_cayley_pack<<<dim3(4), dim3(256), 0, stream>>>(oft_r, Bws);

    // 2) Streaming WMMA transform.
    //    grid.x: 4096 features / (8 waves * 16 feat) = 32
    //    grid.y: 16384 rows / (16 tiles * 16 rows)   = 64
    oft_apply_wmma<<<dim3(32, 64), dim3(256), 0, stream>>>(x, Bws, out);
    (void)in_sizes; (void)n_in; (void)out_size; (void)ws_size;
}